// MatrixApply_65635690218318
// MI455X (gfx1250) — compile-verified
//
#include <hip/hip_runtime.h>

// Block-diagonal GEMM: out[start+r, n] = sum_h M_g[r,h] * x[start+h, n]
// for 15 groups; N = 512*21 = 10752 fp32 columns. All group sizes are
// multiples of 16 -> exact 16x16 tiling for V_WMMA_F32_16X16X4_F32.
//
// Per block (256 thr = 8 waves): one 16-row tile of a group matrix staged in
// LDS, 256 output columns. Each wave owns two 16-col tiles (N-blocking x2)
// so every A fragment feeds two WMMAs.

typedef __attribute__((ext_vector_type(2))) float v2f;
typedef __attribute__((ext_vector_type(8))) float v8f;

#define NCOL 10752            // 512 * 21
#define NGRP 15
#define LDS_STRIDE (320 + 2)  // max group + pad (keeps 8B align, kills bank conflicts)

struct MatPtrs { const float* m[NGRP]; };

__global__ __launch_bounds__(256)
void MatrixApply_wmma_kernel(const float* __restrict__ x,
                             float* __restrict__ out,
                             MatPtrs mats) {
    constexpr int GS[NGRP] = {64, 128, 256, 96, 160, 224, 192, 288, 320,
                              112, 80, 48, 32, 16, 32};

    __shared__ float lds_a[16 * LDS_STRIDE];   // 16 rows of A stripe, padded

    // ---- map blockIdx.y -> (group, row-tile within group, sample start) ----
    int rt = blockIdx.y;           // global 16-row tile index (0..139)
    int start = 0;                 // first sample row of this group
    int g = 16;                    // group size (K dimension)
    int grp = 0;
    #pragma unroll
    for (int i = 0; i < NGRP; ++i) {
        int t = GS[i] >> 4;        // row tiles in this group
        if (rt < t) { g = GS[i]; grp = i; break; }
        rt -= t;
        start += GS[i];
    }
    const float* __restrict__ m = mats.m[grp];
    const int row0 = rt << 4;      // first row of tile within group

    const int tid  = threadIdx.x;
    const int lane = tid & 31;
    const int wv   = tid >> 5;     // wave id 0..7
    const int lo   = lane & 15;    // M (A) / N (B,C) position
    const int hi   = lane >> 4;    // selects K pair {0,1} vs {2,3}

    // ---- stage A stripe (rows row0..row0+15, all g columns) into LDS ----
    // lds[r*(g+2) + c] == lds[i + 2*(i/g)] for i = r*g + c
    {
        const float* __restrict__ ms = m + row0 * g;
        const int tot = g << 4;
        for (int i = tid; i < tot; i += 256) {
            lds_a[i + 2 * (i / g)] = ms[i];
        }
    }
    __syncthreads();

    // ---- per-wave output tiles: 16 rows x {16 cols, 16 cols @ +128} ----
    const int ncol = blockIdx.x * 256 + wv * 16 + lo;
    const float* __restrict__ xb = x + (size_t)start * NCOL + ncol;
    float* __restrict__ ob = out + ((size_t)start + row0) * NCOL + ncol;

    // A fragment source in LDS: row = lo, K offset = 2*hi
    const float* arow = &lds_a[lo * LDS_STRIDE + 2 * hi];

    v8f c0 = {};
    v8f c1 = {};
    for (int k0 = 0; k0 < g; k0 += 16) {
        // A fragments (ds_load_2addr_b64 pairs): K pairs for the 4 K-steps
        v2f a0 = *(const v2f*)(arow + k0);
        v2f a1 = *(const v2f*)(arow + k0 + 4);
        v2f a2 = *(const v2f*)(arow + k0 + 8);
        v2f a3 = *(const v2f*)(arow + k0 + 12);

        // B fragments: row = start + k0 + 4j + 2*hi (+1), cols ncol, ncol+128
        const float* xr = xb + (size_t)(k0 + 2 * hi) * NCOL;
        __builtin_prefetch(xr + (size_t)16 * NCOL, 0, 1);        // global_prefetch_b8
        __builtin_prefetch(xr + (size_t)16 * NCOL + 128, 0, 1);

        v2f b00, b01, b02, b03, b10, b11, b12, b13;
        b00.x = xr[0];            b00.y = xr[NCOL];
        b01.x = xr[4 * NCOL];     b01.y = xr[5 * NCOL];
        b02.x = xr[8 * NCOL];     b02.y = xr[9 * NCOL];
        b03.x = xr[12 * NCOL];    b03.y = xr[13 * NCOL];
        b10.x = xr[128];          b10.y = xr[NCOL + 128];
        b11.x = xr[4 * NCOL + 128]; b11.y = xr[5 * NCOL + 128];
        b12.x = xr[8 * NCOL + 128]; b12.y = xr[9 * NCOL + 128];
        b13.x = xr[12 * NCOL + 128]; b13.y = xr[13 * NCOL + 128];

        // Each A fragment feeds two WMMAs (two N tiles)
        c0 = __builtin_amdgcn_wmma_f32_16x16x4_f32(false, a0, false, b00,
                                                   (short)0, c0, false, false);
        c1 = __builtin_amdgcn_wmma_f32_16x16x4_f32(false, a0, false, b10,
                                                   (short)0, c1, false, false);
        c0 = __builtin_amdgcn_wmma_f32_16x16x4_f32(false, a1, false, b01,
                                                   (short)0, c0, false, false);
        c1 = __builtin_amdgcn_wmma_f32_16x16x4_f32(false, a1, false, b11,
                                                   (short)0, c1, false, false);
        c0 = __builtin_amdgcn_wmma_f32_16x16x4_f32(false, a2, false, b02,
                                                   (short)0, c0, false, false);
        c1 = __builtin_amdgcn_wmma_f32_16x16x4_f32(false, a2, false, b12,
                                                   (short)0, c1, false, false);
        c0 = __builtin_amdgcn_wmma_f32_16x16x4_f32(false, a3, false, b03,
                                                   (short)0, c0, false, false);
        c1 = __builtin_amdgcn_wmma_f32_16x16x4_f32(false, a3, false, b13,
                                                   (short)0, c1, false, false);
    }

    // ---- store C: VGPR vi -> row row0 + vi + 8*hi, cols ncol / ncol+128 ----
    #pragma unroll
    for (int vi = 0; vi < 8; ++vi) {
        ob[(size_t)(vi + 8 * hi) * NCOL] = c0[vi];
        ob[(size_t)(vi + 8 * hi) * NCOL + 128] = c1[vi];
    }
}

extern "C" void kernel_launch(void* const* d_in, const int* in_sizes, int n_in,
                              void* d_out, int out_size, void* d_ws, size_t ws_size,
                              hipStream_t stream) {
    (void)in_sizes; (void)n_in; (void)d_ws; (void)ws_size; (void)out_size;

    const float* x = (const float*)d_in[0];
    float* out = (float*)d_out;

    MatPtrs mats;
    for (int i = 0; i < NGRP; ++i) mats.m[i] = (const float*)d_in[1 + i];

    // 10752 cols / (8 waves * 32 cols) = 42 ; sum(GS)/16 = 128 row tiles
    dim3 grid(42, 140, 1);
    dim3 block(256, 1, 1);
    MatrixApply_wmma_kernel<<<grid, block, 0, stream>>>(x, out, mats);
}